// MSSSIMLoss_19353122636182
// MI455X (gfx1250) — compile-verified
//
#include <hip/hip_runtime.h>
#include <math.h>

typedef float v2f __attribute__((ext_vector_type(2)));
typedef float v8f __attribute__((ext_vector_type(8)));

#define WAVES   4
#define TS      16
#define HALO    26      // TS + 10 (11-tap window)
#define HSTRIDE 27      // padded halo row stride
#define KPAD    28      // K padded to 7 chunks of 4

// ---------------------------------------------------------------------------
// SSIM map + partial reduction. One wave32 per 16x16 output tile.
// Separable 11x11 Gaussian: horizontal pass on VALU, vertical pass as
// Out(16x16) = Toeplitz(16x28) x H(28x16) via V_WMMA_F32_16X16X4_F32.
// Filtered maps stored K-contiguous ([n][k]) so each B chunk is one
// aligned ds_load_b64; Toeplitz A taps precomputed branch-free.
// ---------------------------------------------------------------------------
__global__ __launch_bounds__(128) void ssim_tile_kernel(
    const float* __restrict__ pred, const float* __restrict__ targ,
    int H, int W, int numTiles, int last, float* __restrict__ partials)
{
    __shared__ float sIn[WAVES][2][HALO * HSTRIDE];  // pred / target halos
    __shared__ float sHT[WAVES][5][TS * KPAD];       // filtered maps, [n][k] layout
    __shared__ float sWaveSum[WAVES];
    __shared__ float sG[16];                         // gaussian taps

    const int tid  = threadIdx.x;
    const int wid  = tid >> 5;
    const int lane = tid & 31;

    // Normalized gaussian taps (sigma = 1.5, K = 11), matches reference.
    if (tid == 0) {
        float tmp[11]; float s = 0.f;
        for (int j = 0; j < 11; ++j) {
            float x = (float)(j - 5);
            tmp[j] = expf(-(x * x) / 4.5f);
            s += tmp[j];
        }
        for (int j = 0; j < 11; ++j) sG[j] = tmp[j] / s;
    }
    __syncthreads();

    // Cache taps in registers (constant indices -> stays in VGPRs).
    float gr[11];
#pragma unroll
    for (int j = 0; j < 11; ++j) gr[j] = sG[j];

    const int tilesX  = W >> 4;
    const int tilesY  = H >> 4;
    const int tileIdx = blockIdx.x * WAVES + wid;
    const bool valid  = (tileIdx < numTiles);

    if (valid) {
        int tpi = tilesX * tilesY;
        int img = tileIdx / tpi;
        int rem = tileIdx - img * tpi;
        int ty  = rem / tilesX;
        int tx  = rem - ty * tilesX;
        int y0 = ty * TS, x0 = tx * TS;
        const size_t imgOff = (size_t)img * H * W;

        // ---- halo load (26x26, zero-padded at image borders) ----
        for (int i = lane; i < HALO * HALO; i += 32) {
            int r = i / HALO, c = i - r * HALO;
            int y = y0 - 5 + r, x = x0 - 5 + c;
            bool in = (y >= 0) & (y < H) & (x >= 0) & (x < W);
            size_t a = imgOff + (size_t)y * W + x;
            sIn[wid][0][r * HSTRIDE + c] = in ? pred[a] : 0.f;
            sIn[wid][1][r * HSTRIDE + c] = in ? targ[a] : 0.f;
        }
    }
    __syncthreads();

    // ---- horizontal 11-tap pass, fused p^2 / t^2 / p*t ----
    // Output stored transposed: sHT[q][c * KPAD + r]; rows 26..27 zeroed.
    if (valid) {
        for (int i = lane; i < KPAD * TS; i += 32) {
            int r = i >> 4, c = i & 15;          // r = K index, c = column
            float ap = 0.f, at = 0.f, app = 0.f, att = 0.f, apt = 0.f;
            if (r < HALO) {
                const float* pRow = &sIn[wid][0][r * HSTRIDE + c];
                const float* tRow = &sIn[wid][1][r * HSTRIDE + c];
#pragma unroll
                for (int j = 0; j < 11; ++j) {
                    float gw = gr[j];
                    float p = pRow[j], t = tRow[j];
                    ap  += gw * p;      at  += gw * t;
                    app += gw * p * p;  att += gw * t * t;  apt += gw * p * t;
                }
            }
            int o = c * KPAD + r;
            sHT[wid][0][o] = ap;  sHT[wid][1][o] = at;
            sHT[wid][2][o] = app; sHT[wid][3][o] = att; sHT[wid][4][o] = apt;
        }
    }
    __syncthreads();

    // ---- vertical pass as WMMA: D(16x16) += A(16x4) * B(4x16), 7 K-chunks ----
    float acc = 0.f;
    if (valid) {
        const int rA   = lane & 15;   // A: M = lane%16
        const int half = lane >> 4;
        const int n    = lane & 15;   // B/D: N = lane%16

        // Precompute Toeplitz A taps for all 7 chunks, branch-free:
        // T[m,k] = g[k-m] for 0 <= k-m <= 10, else 0.
        float a0r[7], a1r[7];
#pragma unroll
        for (int kc = 0; kc < 7; ++kc) {
            int k0 = 4 * kc + 2 * half;
            unsigned d0 = (unsigned)(k0 - rA);
            unsigned d1 = (unsigned)(k0 + 1 - rA);
            float t0 = sG[d0 <= 10u ? d0 : 10u];   // unconditional clamped load
            float t1 = sG[d1 <= 10u ? d1 : 10u];
            a0r[kc] = (d0 <= 10u) ? t0 : 0.f;      // v_cndmask, no exec branch
            a1r[kc] = (d1 <= 10u) ? t1 : 0.f;
        }

        v8f d[5];
#pragma unroll
        for (int q = 0; q < 5; ++q)
#pragma unroll
            for (int e = 0; e < 8; ++e) d[q][e] = 0.f;

        const int bBase = n * KPAD + 2 * half;     // element index of (k0) for kc=0
#pragma unroll
        for (int kc = 0; kc < 7; ++kc) {
            v2f a; a[0] = a0r[kc]; a[1] = a1r[kc];
            const int bo = bBase + 4 * kc;         // even -> 8B aligned pair
#pragma unroll
            for (int q = 0; q < 5; ++q) {
                v2f b = *(const v2f*)&sHT[wid][q][bo];   // ds_load_b64
                d[q] = __builtin_amdgcn_wmma_f32_16x16x4_f32(
                    false, a, false, b, (short)0, d[q], false, false);
            }
        }

        // ---- per-pixel SSIM epilogue (8 pixels per lane, D layout) ----
        const float C1 = 1e-4f, C2 = 9e-4f;
#pragma unroll
        for (int e = 0; e < 8; ++e) {
            float mp  = d[0][e], mt = d[1][e];
            float mpp = mp * mp, mtt = mt * mt, mpt = mp * mt;
            float spp = d[2][e] - mpp;
            float stt = d[3][e] - mtt;
            float spt = d[4][e] - mpt;
            float cs  = (2.f * spt + C2) / (spp + stt + C2);
            float v   = cs;
            if (last) v *= (2.f * mpt + C1) / (mpp + mtt + C1);
            acc += v;
        }
    }

    // ---- wave reduction (wave32!) then block partial ----
    for (int m = 16; m > 0; m >>= 1)
        acc += __shfl_xor(acc, m, 32);
    if (lane == 0) sWaveSum[wid] = acc;
    __syncthreads();
    if (tid == 0)
        partials[blockIdx.x] = sWaveSum[0] + sWaveSum[1] + sWaveSum[2] + sWaveSum[3];
}

// Deterministic fixed-order mean of per-block partials.
__global__ void reduce_mean_kernel(const float* __restrict__ partials, int n,
                                   float invCount, float* __restrict__ outSum)
{
    __shared__ float s[256];
    int t = threadIdx.x;
    float a = 0.f;
    for (int i = t; i < n; i += 256) a += partials[i];
    s[t] = a;
    __syncthreads();
    for (int m = 128; m > 0; m >>= 1) {
        if (t < m) s[t] += s[t + m];
        __syncthreads();
    }
    if (t == 0) *outSum = s[0] * invCount;
}

// 2x2 average pool, NCHW flattened as (48, H, W).
__global__ void downsample_kernel(const float* __restrict__ in, float* __restrict__ out,
                                  int H2, int W2, int total)
{
    int i = blockIdx.x * blockDim.x + threadIdx.x;
    if (i >= total) return;
    int per = H2 * W2;
    int img = i / per;
    int rem = i - img * per;
    int y = rem / W2;
    int x = rem - y * W2;
    int W = W2 * 2;
    const float* base = in + (size_t)img * (H2 * 2) * W + (size_t)(2 * y) * W + 2 * x;
    out[i] = 0.25f * (base[0] + base[1] + base[W] + base[W + 1]);
}

// out = 1 - s4 * prod_{i<4} s_i^{w_i}   (reference applies no exponent to last)
__global__ void finalize_kernel(const float* __restrict__ sums, float* __restrict__ out)
{
    if (threadIdx.x == 0 && blockIdx.x == 0) {
        const float w[4] = {0.0448f, 0.2856f, 0.3001f, 0.2363f};
        float m = sums[4];
        for (int i = 0; i < 4; ++i) m *= powf(sums[i], w[i]);
        out[0] = 1.f - m;
    }
}

extern "C" void kernel_launch(void* const* d_in, const int* in_sizes, int n_in,
                              void* d_out, int out_size, void* d_ws, size_t ws_size,
                              hipStream_t stream)
{
    (void)in_sizes; (void)n_in; (void)out_size; (void)ws_size;

    const float* pred0 = (const float*)d_in[0];
    const float* targ0 = (const float*)d_in[1];
    float* ws = (float*)d_ws;

    // Workspace layout: pyramid levels 1..4 (pred+target), partials, 5 level sums.
    const float* predL[5]; const float* targL[5];
    float* predW[5] = {nullptr}; float* targW[5] = {nullptr};
    predL[0] = pred0; targL[0] = targ0;
    size_t off = 0;
    int H = 512;
    for (int l = 1; l < 5; ++l) {
        H >>= 1;
        size_t n = (size_t)48 * H * H;
        predW[l] = ws + off; off += n;
        targW[l] = ws + off; off += n;
        predL[l] = predW[l]; targL[l] = targW[l];
    }
    float* partials = ws + off; off += 12288;   // max blocks at level 0
    float* sums     = ws + off; off += 8;

    H = 512;
    for (int l = 0; l < 5; ++l) {
        int tiles = 48 * (H / 16) * (H / 16);
        int nb = (tiles + WAVES - 1) / WAVES;
        ssim_tile_kernel<<<nb, 128, 0, stream>>>(predL[l], targL[l], H, H, tiles,
                                                 (l == 4) ? 1 : 0, partials);
        float invCount = 1.0f / (48.0f * (float)H * (float)H);
        reduce_mean_kernel<<<1, 256, 0, stream>>>(partials, nb, invCount, sums + l);
        if (l < 4) {
            int H2 = H / 2;
            int tot = 48 * H2 * H2;
            int blk = (tot + 255) / 256;
            downsample_kernel<<<blk, 256, 0, stream>>>(predL[l], predW[l + 1], H2, H2, tot);
            downsample_kernel<<<blk, 256, 0, stream>>>(targL[l], targW[l + 1], H2, H2, tot);
            H = H2;
        }
    }
    finalize_kernel<<<1, 1, 0, stream>>>(sums, (float*)d_out);
}